// SetAbstraction_31336081392126
// MI455X (gfx1250) — compile-verified
//
#include <hip/hip_runtime.h>
#include <hip/hip_bf16.h>
#include <stdint.h>

// ---------------------------------------------------------------------------
// PointNet++ SetAbstraction for MI455X (gfx1250, wave32).
// Phases: FPS -> ball-query+gather -> [WMMA f32 GEMM -> BN stats -> norm+ReLU] x3
//         -> max-over-nsample.
// GEMM: V_WMMA_F32_16X16X4_F32, weights staged into LDS via the Tensor Data
// Mover (tensor_load_to_lds + s_wait_tensorcnt), A strip preloaded to VGPRs.
// ---------------------------------------------------------------------------

#define B_      8
#define N_      8192
#define C_IN    64
#define NPOINT  2048
#define NSAMPLE 32
#define RAD2    0.04f       // 0.2^2
#define K0PAD   68          // 3 + 64 = 67 padded to multiple of 4
#define MROWS   ((size_t)B_ * NPOINT * NSAMPLE)   // 524288
#define EPSBN   1e-5f

typedef __attribute__((ext_vector_type(2))) float v2f;
typedef __attribute__((ext_vector_type(8))) float v8f;
typedef __attribute__((ext_vector_type(4))) unsigned int u32x4;
typedef __attribute__((ext_vector_type(8))) int i32x8;
typedef __attribute__((ext_vector_type(4))) int i32x4;

#if __has_builtin(__builtin_amdgcn_ballot_w32)
#define BALLOT32(p) __builtin_amdgcn_ballot_w32(p)
#else
#define BALLOT32(p) ((unsigned)__ballot(p))
#endif

#if __has_builtin(__builtin_amdgcn_tensor_load_to_lds) && \
    __has_builtin(__builtin_amdgcn_s_wait_tensorcnt)
#define USE_TDM 1
#else
#define USE_TDM 0
#endif

// ---------------------------------------------------------------------------
// 1) Farthest point sampling: one block (1024 thr) per batch; 8 pts/thread in
//    registers; per-step two-level argmax (shfl within wave32, LDS across).
// ---------------------------------------------------------------------------
__global__ __launch_bounds__(1024)
void fps_kernel(const float* __restrict__ xyz, const int* __restrict__ start_idx,
                float* __restrict__ new_xyz /* == d_out head */) {
    const int b = blockIdx.x;
    const int t = threadIdx.x;
    const float* base = xyz + (size_t)b * N_ * 3;

    float px[8], py[8], pz[8], dd[8];
#pragma unroll
    for (int i = 0; i < 8; ++i) {
        int p = t * 8 + i;
        px[i] = base[p * 3 + 0];
        py[i] = base[p * 3 + 1];
        pz[i] = base[p * 3 + 2];
        dd[i] = 1e10f;
    }

    __shared__ float bc[3];
    __shared__ float wv[32];
    __shared__ int   wi[32];
    __shared__ int   sCur;

    int far = start_idx[b];
    for (int iter = 0; iter < NPOINT; ++iter) {
        if (t == (far >> 3)) {                    // owner broadcasts centroid
            int j = far & 7;
            bc[0] = px[j]; bc[1] = py[j]; bc[2] = pz[j];
            float* o = new_xyz + ((size_t)b * NPOINT + iter) * 3;
            o[0] = px[j]; o[1] = py[j]; o[2] = pz[j];
        }
        __syncthreads();
        float cx = bc[0], cy = bc[1], cz = bc[2];

        float best = -1.0f; int bi = 0;
#pragma unroll
        for (int i = 0; i < 8; ++i) {
            float dx = px[i] - cx, dy = py[i] - cy, dz = pz[i] - cz;
            float d = dx * dx + dy * dy + dz * dz;
            dd[i] = fminf(dd[i], d);
            if (dd[i] > best) { best = dd[i]; bi = t * 8 + i; }
        }
        // wave32 reduce (keeps lowest index on ties -> matches argmax)
#pragma unroll
        for (int off = 16; off > 0; off >>= 1) {
            float ov = __shfl_down(best, off, 32);
            int   oi = __shfl_down(bi, off, 32);
            if (ov > best) { best = ov; bi = oi; }
        }
        int lane = t & 31, wave = t >> 5;
        if (lane == 0) { wv[wave] = best; wi[wave] = bi; }
        __syncthreads();
        if (wave == 0) {
            float v = wv[lane]; int i2 = wi[lane];
#pragma unroll
            for (int off = 16; off > 0; off >>= 1) {
                float ov = __shfl_down(v, off, 32);
                int   oi = __shfl_down(i2, off, 32);
                if (ov > v) { v = ov; i2 = oi; }
            }
            if (lane == 0) sCur = i2;
        }
        __syncthreads();
        far = sCur;
    }
}

// ---------------------------------------------------------------------------
// 2) Ball query (first NSAMPLE in index order inside radius) + gather + concat
//    into x0[M, 68].  One wave per centroid; in-order selection via ballot.
// ---------------------------------------------------------------------------
__global__ __launch_bounds__(256)
void ball_gather_kernel(const float* __restrict__ xyz, const float* __restrict__ points,
                        const float* __restrict__ new_xyz, float* __restrict__ x0) {
    const int wave = threadIdx.x >> 5, lane = threadIdx.x & 31;
    const int s = blockIdx.x * 8 + wave;
    const int b = blockIdx.y;

    const float* nx = new_xyz + ((size_t)b * NPOINT + s) * 3;
    const float cx = nx[0], cy = nx[1], cz = nx[2];
    const float* xb = xyz + (size_t)b * N_ * 3;

    int cnt = 0, myIdx = -1;
    for (int basep = 0; basep < N_ && cnt < NSAMPLE; basep += 32) {
        int p = basep + lane;
        float dx = xb[p * 3 + 0] - cx;
        float dy = xb[p * 3 + 1] - cy;
        float dz = xb[p * 3 + 2] - cz;
        float d = dx * dx + dy * dy + dz * dz;
        unsigned mask = BALLOT32(d <= RAD2);
        int pc = __popc(mask);
        if (lane >= cnt && lane < cnt + pc) {      // this lane owns slot `lane`
            unsigned m = mask;
            for (int q = lane - cnt; q > 0; --q) m &= (m - 1);
            myIdx = basep + __builtin_ctz(m);
        }
        cnt += pc;
    }
    int firstIdx = __shfl(myIdx, 0, 32);           // fill empty slots with first
    if (myIdx < 0) myIdx = firstIdx;
    if (myIdx < 0) myIdx = 0;                      // (cannot happen: centroid in ball)

    const float* ps = xb + (size_t)myIdx * 3;
    size_t row = ((size_t)(b * NPOINT + s)) * NSAMPLE + lane;
    float* o = x0 + row * K0PAD;
    o[0] = ps[0] - cx;
    o[1] = ps[1] - cy;
    o[2] = ps[2] - cz;
    const float* pf = points + ((size_t)b * N_ + myIdx) * C_IN;
#pragma unroll 4
    for (int c = 0; c < C_IN; ++c) o[3 + c] = pf[c];
    o[67] = 0.0f;
}

// ---------------------------------------------------------------------------
// 3) WMMA f32 GEMM: Y[M,C] = A[M,K] * W[K,C], K = KCH*4, C = TN*16.
//    - W tile (K x C, <=32KB) staged to LDS once per block via the TDM
//      (tensor_load_to_lds + s_wait_tensorcnt), cooperative-load fallback.
//    - Per wave: full 16-row A strip preloaded to VGPRs (b64 loads), then the
//      MAC loop touches only LDS for B fragments.
//    ISA fragment layouts (f32):
//      A 16x4 : lanes 0-15 row l hold (K0,K1); lanes 16-31 hold (K2,K3)
//      B 4x16 : v0 = rows {K0 | K2}, v1 = rows {K1 | K3} across lane halves
//      C/D    : vgpr r = rows {r | r+8}, col = lane&15
// ---------------------------------------------------------------------------
template<int KCH, int TN>
__global__ __launch_bounds__(256)
void gemm_wmma_kernel(const float* __restrict__ A, const float* __restrict__ W,
                      float* __restrict__ Y) {
    constexpr int Kdim = KCH * 4;
    constexpr int Cdim = TN * 16;
    __shared__ float shW[Kdim * Cdim];

#if USE_TDM
    if (threadIdx.x == 0) {
        // Tensor DMA descriptor for the full 2D tile: Kdim rows x Cdim cols, f32.
        uint64_t ga  = (uint64_t)(uintptr_t)W;
        uint32_t lds = (uint32_t)(uintptr_t)shW;   // generic addr low 32b = LDS offset
        u32x4 g0;
        g0[0] = 1u;                                             // count=1 (valid)
        g0[1] = lds;                                            // lds_addr
        g0[2] = (uint32_t)ga;                                   // global_addr[31:0]
        g0[3] = (uint32_t)((ga >> 32) & 0x01FFFFFFu) | (2u << 30); // addr[56:32] | type=2
        i32x8 g1;
        g1[0] = (int)(2u << 16);                 // wg_mask=0 | data_size=2 (4B)
        g1[1] = (int)((unsigned)Cdim << 16);     // tensor_dim0[15:0] = Cdim
        g1[2] = (int)((unsigned)Kdim << 16);     // dim0[31:16]=0 | tensor_dim1[15:0]=Kdim
        g1[3] = (int)((unsigned)Cdim << 16);     // dim1[31:16]=0 | tile_dim0 = Cdim
        g1[4] = (int)(unsigned)Kdim;             // tile_dim1 = Kdim | tile_dim2 = 0
        g1[5] = (int)(unsigned)Cdim;             // tensor_dim0_stride[31:0] = Cdim
        g1[6] = 0;                               // stride0[47:32] | stride1[15:0]
        g1[7] = 0;                               // stride1[47:16]
        i32x4 z4 = {0, 0, 0, 0};
        i32x8 z8 = {0, 0, 0, 0, 0, 0, 0, 0};
        __builtin_amdgcn_tensor_load_to_lds(g0, g1, z4, z4, z8, 0);
        __builtin_amdgcn_s_wait_tensorcnt(0);
    }
    __syncthreads();
#else
    for (int i = threadIdx.x; i < Kdim * Cdim; i += 256) shW[i] = W[i];
    __syncthreads();
#endif

    const int wave = threadIdx.x >> 5;
    const int lane = threadIdx.x & 31;
    const size_t row0 = ((size_t)blockIdx.x * 8 + wave) * 16;
    const int half = lane >> 4;
    const int l = lane & 15;

    // Preload the whole A strip: 2 f32 per lane per k-chunk (aligned b64 loads).
    const float* Arow = A + (row0 + (size_t)l) * (size_t)Kdim + 2 * half;
    v2f areg[KCH];
#pragma unroll
    for (int kc = 0; kc < KCH; ++kc)
        areg[kc] = *(const v2f*)(Arow + kc * 4);

    v8f acc[TN];
#pragma unroll
    for (int t = 0; t < TN; ++t) acc[t] = (v8f){0.f,0.f,0.f,0.f,0.f,0.f,0.f,0.f};

    // Per-lane LDS base; the (kc, t) part is a uniform immediate DS offset.
    const float* wbase = shW + 2 * half * Cdim + l;
#pragma unroll
    for (int kc = 0; kc < KCH; ++kc) {
#pragma unroll
        for (int t = 0; t < TN; ++t) {
            v2f bv;
            bv.x = wbase[kc * 4 * Cdim + t * 16];
            bv.y = wbase[kc * 4 * Cdim + t * 16 + Cdim];
            acc[t] = __builtin_amdgcn_wmma_f32_16x16x4_f32(
                false, areg[kc], false, bv, (short)0, acc[t], false, false);
        }
    }
#pragma unroll
    for (int t = 0; t < TN; ++t) {
        float* o = Y + (row0 + 8 * (size_t)half) * (size_t)Cdim + t * 16 + l;
#pragma unroll
        for (int r = 0; r < 8; ++r) o[(size_t)r * Cdim] = acc[t][r];
    }
}

// ---------------------------------------------------------------------------
// 4) BN statistics: deterministic two-pass per-channel sum / sumsq.
// ---------------------------------------------------------------------------
__global__ __launch_bounds__(256)
void stats_part_kernel(const float* __restrict__ y, size_t M, int C,
                       float* __restrict__ part, float* __restrict__ part2) {
    __shared__ float ls[256], ls2[256];
    const int tid = threadIdx.x;
    const int c = tid & (C - 1);
    const int g = tid / C;
    const int rpi = 256 / C;
    float s = 0.f, s2 = 0.f;
    for (size_t r = (size_t)blockIdx.x * rpi + g; r < M; r += (size_t)gridDim.x * rpi) {
        float v = y[r * (size_t)C + c];
        s += v; s2 += v * v;
    }
    ls[tid] = s; ls2[tid] = s2;
    __syncthreads();
    if (tid < C) {
        for (int j = 1; j < rpi; ++j) { s += ls[j * C + tid]; s2 += ls2[j * C + tid]; }
        part [blockIdx.x * C + tid] = s;
        part2[blockIdx.x * C + tid] = s2;
    }
}

__global__ __launch_bounds__(128)
void stats_reduce_kernel(const float* __restrict__ part, const float* __restrict__ part2,
                         int C, int nb, float* __restrict__ sum, float* __restrict__ sumsq) {
    const int c = threadIdx.x;
    if (c >= C) return;
    float s = 0.f, s2 = 0.f;
    for (int j = 0; j < nb; ++j) { s += part[j * C + c]; s2 += part2[j * C + c]; }
    sum[c] = s; sumsq[c] = s2;
}

// ---------------------------------------------------------------------------
// 5) Normalize + ReLU in place (layers 1,2).  Conv bias cancels exactly under
//    BN mean subtraction, so it is intentionally omitted.
// ---------------------------------------------------------------------------
__global__ __launch_bounds__(256)
void norm_relu_kernel(float* __restrict__ y, size_t M, int C,
                      const float* __restrict__ sum, const float* __restrict__ sumsq,
                      const float* __restrict__ gamma, const float* __restrict__ beta) {
    const size_t total = M * (size_t)C;
    const float invM = 1.0f / (float)M;
    for (size_t i = (size_t)blockIdx.x * blockDim.x + threadIdx.x; i < total;
         i += (size_t)gridDim.x * blockDim.x) {
        int c = (int)(i & (size_t)(C - 1));
        float mean = sum[c] * invM;
        float var = sumsq[c] * invM - mean * mean;
        float v = (y[i] - mean) * rsqrtf(var + EPSBN) * gamma[c] + beta[c];
        y[i] = v > 0.f ? v : 0.f;
    }
}

// ---------------------------------------------------------------------------
// 6) Layer-3 normalize + ReLU fused with max over NSAMPLE -> new_points.
// ---------------------------------------------------------------------------
__global__ __launch_bounds__(128)
void final_max_kernel(const float* __restrict__ y3,
                      const float* __restrict__ sum, const float* __restrict__ sumsq,
                      const float* __restrict__ gamma, const float* __restrict__ beta,
                      float* __restrict__ out) {
    const int c = threadIdx.x;            // 128 channels
    const size_t s = blockIdx.x;          // B*NPOINT groups
    const float invM = 1.0f / (float)MROWS;
    const float mean = sum[c] * invM;
    const float rs = rsqrtf(sumsq[c] * invM - mean * mean + EPSBN);
    const float g = gamma[c], be = beta[c];
    const float* p = y3 + s * NSAMPLE * 128 + c;
    float m = 0.0f;                       // relu() >= 0, so 0 is the identity
#pragma unroll
    for (int k = 0; k < NSAMPLE; ++k) {
        float v = (p[(size_t)k * 128] - mean) * rs * g + be;
        v = v > 0.f ? v : 0.f;
        m = fmaxf(m, v);
    }
    out[s * 128 + c] = m;
}

// ---------------------------------------------------------------------------
// 7) Pad W0 (67x64) -> (68x64) with a zero row to match K0PAD.
// ---------------------------------------------------------------------------
__global__ __launch_bounds__(256)
void pad_w0_kernel(const float* __restrict__ W0, float* __restrict__ W0p) {
    for (int i = threadIdx.x + blockIdx.x * blockDim.x; i < K0PAD * 64;
         i += blockDim.x * gridDim.x) {
        int r = i / 64;
        W0p[i] = (r < 67) ? W0[i] : 0.0f;
    }
}

// ---------------------------------------------------------------------------
// Host-side launcher.
// d_in: 0 xyz, 1 points, 2 start_idx, 3 W0, 4 b0, 5 g0, 6 be0,
//       7 W1, 8 b1, 9 g1, 10 be1, 11 W2, 12 b2, 13 g2, 14 be2
// d_out: [new_xyz (B*NPOINT*3) | new_points (B*NPOINT*128)] float32
// ---------------------------------------------------------------------------
extern "C" void kernel_launch(void* const* d_in, const int* in_sizes, int n_in,
                              void* d_out, int out_size, void* d_ws, size_t ws_size,
                              hipStream_t stream) {
    (void)in_sizes; (void)n_in; (void)out_size; (void)ws_size;

    const float* xyz    = (const float*)d_in[0];
    const float* points = (const float*)d_in[1];
    const int*   start  = (const int*)d_in[2];
    const float* W0 = (const float*)d_in[3];
    const float* g0 = (const float*)d_in[5];
    const float* be0= (const float*)d_in[6];
    const float* W1 = (const float*)d_in[7];
    const float* g1 = (const float*)d_in[9];
    const float* be1= (const float*)d_in[10];
    const float* W2 = (const float*)d_in[11];
    const float* g2 = (const float*)d_in[13];
    const float* be2= (const float*)d_in[14];

    float* out_xyz = (float*)d_out;                                 // B*NPOINT*3
    float* out_np  = (float*)d_out + (size_t)B_ * NPOINT * 3;       // B*NPOINT*128

    // Workspace layout (bytes)
    uint8_t* ws = (uint8_t*)d_ws;
    float* W0p   = (float*)(ws + 0);                    //  68*64*4      = 17408
    float* sums  = (float*)(ws + 17408);                //  6*128*4      =  3072
    float* part  = (float*)(ws + 20480);                //  1024*128*4   (pass-1 sums)
    float* part2 = part + 1024 * 128;                   //  1024*128*4   (pass-1 sumsq)
    float* x0    = (float*)(ws + 1069056);              //  M*68*4  = 142606336
    float* y1    = (float*)(ws + 143675392ull);         //  M*64*4  = 134217728
    float* y2    = x0;                                  //  reuse (x0 dead after GEMM1)
    float* y3    = (float*)(ws + 277893120ull);         //  M*128*4 = 268435456

    float* sum1 = sums + 0,   *sq1 = sums + 128;
    float* sum2 = sums + 256, *sq2 = sums + 384;
    float* sum3 = sums + 512, *sq3 = sums + 640;

    const int STAT_BLOCKS = 1024;
    const int gemm_blocks = (int)(MROWS / (16 * 8));    // 8 wave-tiles per block

    // 0) pad W0 to K=68
    pad_w0_kernel<<<4, 256, 0, stream>>>(W0, W0p);
    // 1) FPS -> new_xyz (directly into d_out)
    fps_kernel<<<B_, 1024, 0, stream>>>(xyz, start, out_xyz);
    // 2) ball query + gather + concat -> x0 [M, 68]
    ball_gather_kernel<<<dim3(NPOINT / 8, B_), 256, 0, stream>>>(xyz, points, out_xyz, x0);

    // Layer 1: K=68 -> C=64
    gemm_wmma_kernel<17, 4><<<gemm_blocks, 256, 0, stream>>>(x0, W0p, y1);
    stats_part_kernel<<<STAT_BLOCKS, 256, 0, stream>>>(y1, MROWS, 64, part, part2);
    stats_reduce_kernel<<<1, 128, 0, stream>>>(part, part2, 64, STAT_BLOCKS, sum1, sq1);
    norm_relu_kernel<<<4096, 256, 0, stream>>>(y1, MROWS, 64, sum1, sq1, g0, be0);

    // Layer 2: K=64 -> C=64
    gemm_wmma_kernel<16, 4><<<gemm_blocks, 256, 0, stream>>>(y1, W1, y2);
    stats_part_kernel<<<STAT_BLOCKS, 256, 0, stream>>>(y2, MROWS, 64, part, part2);
    stats_reduce_kernel<<<1, 128, 0, stream>>>(part, part2, 64, STAT_BLOCKS, sum2, sq2);
    norm_relu_kernel<<<4096, 256, 0, stream>>>(y2, MROWS, 64, sum2, sq2, g1, be1);

    // Layer 3: K=64 -> C=128 (normalize fused into max-pool)
    gemm_wmma_kernel<16, 8><<<gemm_blocks, 256, 0, stream>>>(y2, W2, y3);
    stats_part_kernel<<<STAT_BLOCKS, 256, 0, stream>>>(y3, MROWS, 128, part, part2);
    stats_reduce_kernel<<<1, 128, 0, stream>>>(part, part2, 128, STAT_BLOCKS, sum3, sq3);
    final_max_kernel<<<B_ * NPOINT, 128, 0, stream>>>(y3, sum3, sq3, g2, be2, out_np);
}